// ExcitationGeneration_42855183680039
// MI455X (gfx1250) — compile-verified
//
#include <hip/hip_runtime.h>
#include <math.h>
#include <stdint.h>

// ---------------- problem geometry (fixed by the reference) ----------------
#define P_FRAME          80
#define N_FRAMES         8192
#define B_ROWS           64
#define S_ROW            (N_FRAMES * P_FRAME)          // 655360 samples / row
#define CHUNKS           32                            // blocks per row
#define WPB              8                             // wave32 waves per block
#define BLOCK_T          (WPB * 32)                    // 256 threads
#define WAVE_TILE        (S_ROW / (CHUNKS * WPB))      // 2560 samples / wave
#define STEPS            (WAVE_TILE / 32)              // 80 steps of 32 lanes
#define BLOCK_TILE       (WAVE_TILE * WPB)             // 20480 samples / block
#define FRAMES_PER_BLOCK (BLOCK_TILE / P_FRAME)        // 256 frames / block
#define LDS_F            (FRAMES_PER_BLOCK + 2)        // f0-1 .. f0+256
#define WAVES_PER_ROW    (CHUNKS * WPB)                // 256 partials / row

// ---------------- CDNA5 async global->LDS path (probe via __has_builtin) ---
#if defined(__has_builtin)
#  if __has_builtin(__builtin_amdgcn_global_load_async_to_lds_b32)
#    define HAVE_ASYNC_LDS 1
#  endif
#  if __has_builtin(__builtin_amdgcn_s_wait_asynccnt)
#    define HAVE_WAIT_ASYNC 1
#  endif
#endif
#ifndef HAVE_ASYNC_LDS
#  define HAVE_ASYNC_LDS 0
#endif
#ifndef HAVE_WAIT_ASYNC
#  define HAVE_WAIT_ASYNC 0
#endif

// prototype (from clang diagnostic): (global int*, shared int*, imm offset, imm cpol)
typedef __attribute__((address_space(1))) int* gptr1_t;
typedef __attribute__((address_space(3))) int* lptr3_t;

// ---------------- wave32 lane-scan helpers ---------------------------------
__device__ __forceinline__ float wave_scan_add(float v, int lane) {
#pragma unroll
  for (int off = 1; off < 32; off <<= 1) {
    float t = __shfl_up(v, off, 32);
    if (lane >= off) v += t;
  }
  return v;
}
__device__ __forceinline__ float wave_scan_max(float v, int lane) {
#pragma unroll
  for (int off = 1; off < 32; off <<= 1) {
    float t = __shfl_up(v, off, 32);
    if (lane >= off) v = fmaxf(v, t);
  }
  return v;
}
__device__ __forceinline__ float wave_reduce_max(float v) {
#pragma unroll
  for (int off = 16; off >= 1; off >>= 1) v = fmaxf(v, __shfl_xor(v, off, 32));
  return v;
}
__device__ __forceinline__ float bcast31(float v) { return __shfl(v, 31, 32); }

// ---------------- stage one block's frame window into LDS ------------------
// ldsP[i]  = p[clamp(f0-1+i)]           (original pitch, for voiced mask)
// ldsPP[i] = hold-corrected pitch p'[f0-1+i]  (valid for i>=1)
// wTab[k]  = k / 80.0f  (exact same rounding as jnp.arange(P)/P)
__device__ __forceinline__ void stage_frames(const float* __restrict__ pRow, int f0,
                                             float* ldsP, float* ldsPP, float* wTab) {
  const int tid = threadIdx.x;
  for (int k = tid; k < P_FRAME; k += BLOCK_T)
    wTab[k] = (float)k / (float)P_FRAME;
  for (int i = tid; i < LDS_F; i += BLOCK_T) {
    int fg = f0 - 1 + i;
    fg = fg < 0 ? 0 : (fg > N_FRAMES - 1 ? N_FRAMES - 1 : fg);
#if HAVE_ASYNC_LDS
    __builtin_amdgcn_global_load_async_to_lds_b32(
        (gptr1_t)(pRow + fg), (lptr3_t)&ldsP[i], 0, 0);
#else
    ldsP[i] = pRow[fg];
#endif
  }
#if HAVE_ASYNC_LDS && HAVE_WAIT_ASYNC
  __builtin_amdgcn_s_wait_asynccnt(0);
#endif
  __syncthreads();
  // hold previous pitch at voiced->unvoiced transitions:
  // p'[f] = (p[f]==0 && f>=1 && p[f-1]>0) ? p[f-1] : p[f]
  for (int i = tid + 1; i < LDS_F; i += BLOCK_T) {
    int fg = f0 - 1 + i;
    fg = fg > N_FRAMES - 1 ? N_FRAMES - 1 : fg;
    float pf = ldsP[i];
    float pm = ldsP[i - 1];
    bool hold = (pf == 0.0f) && (fg >= 1) && (pm > 0.0f);
    ldsPP[i] = hold ? pm : pf;
  }
  __syncthreads();
}

// per-sample: voiced mask, interpolated pitch, instantaneous freq q = 1/p
__device__ __forceinline__ void sample_q(const float* ldsP, const float* ldsPP,
                                         const float* wTab, int nLocal,
                                         bool& mask, float& psamp, float& q) {
  int fL = nLocal / P_FRAME;                 // 0..FRAMES_PER_BLOCK-1
  int k  = nLocal - fL * P_FRAME;
  mask = ldsP[fL + 1] > 0.0f;
  float w    = wTab[k];
  float ppf  = ldsPP[fL + 1];
  float ppf1 = ldsPP[fL + 2];
  float y = ppf * (1.0f - w) + ppf1 * w;     // linear interpolation
  psamp = mask ? y : 0.0f;
  q     = mask ? 1.0f / y : 0.0f;            // y>0 whenever mask (hold guarantees it)
}

// ---------------- kernel 1: per-wave partials (Qsum, M) --------------------
// M = max over unvoiced samples in tile of (tile-local prefix sum), -inf if none
__global__ void __launch_bounds__(BLOCK_T)
exgen_partials(const float* __restrict__ p, float2* __restrict__ part) {
  __shared__ float ldsP[LDS_F];
  __shared__ float ldsPP[LDS_F];
  __shared__ float wTab[P_FRAME];
  const int chunk = blockIdx.x, row = blockIdx.y;
  const float* pRow = p + (size_t)row * N_FRAMES;
  stage_frames(pRow, chunk * FRAMES_PER_BLOCK, ldsP, ldsPP, wTab);

  const int lane = threadIdx.x & 31, wave = threadIdx.x >> 5;
  const float NEG_INF = -__builtin_inff();
  float runS = 0.0f, M = NEG_INF;
  const int base = wave * WAVE_TILE + lane;
  for (int s2 = 0; s2 < STEPS; ++s2) {
    int nLocal = base + s2 * 32;
    bool mask; float psamp, q;
    sample_q(ldsP, ldsPP, wTab, nLocal, mask, psamp, q);
    float sInc = wave_scan_add(q, lane);
    float cand = mask ? NEG_INF : (runS + sInc);   // bias candidate = s at unvoiced
    M = fmaxf(M, wave_reduce_max(cand));
    runS += bcast31(sInc);
  }
  if (lane == 0) {
    int gw = (row * CHUNKS + chunk) * WPB + wave;
    part[gw] = make_float2(runS, M);
  }
}

// ---------------- kernel 2: in-place inclusive pair-scan per row -----------
// combine((S,B),(S',B')) = (S+S', max(B, S+B'))
__global__ void __launch_bounds__(32)
exgen_scan(float2* __restrict__ part) {
  const int lane = threadIdx.x;
  const float NEG_INF = -__builtin_inff();
  float2* base = part + (size_t)blockIdx.x * WAVES_PER_ROW;
  const int PER = WAVES_PER_ROW / 32;               // 8 elements per lane
  float2 v[PER];
  float aS = 0.0f, aB = NEG_INF;
#pragma unroll
  for (int j = 0; j < PER; ++j) {
    v[j] = base[lane * PER + j];
    aB = fmaxf(aB, aS + v[j].y);
    aS += v[j].x;
  }
  float S = aS, Bm = aB;                            // wave inclusive pair-scan
#pragma unroll
  for (int off = 1; off < 32; off <<= 1) {
    float tS = __shfl_up(S, off, 32);
    float tB = __shfl_up(Bm, off, 32);
    if (lane >= off) { Bm = fmaxf(tB, tS + Bm); S = tS + S; }
  }
  float cS = __shfl_up(S, 1, 32);                   // exclusive lane carry
  float cB = __shfl_up(Bm, 1, 32);
  if (lane == 0) { cS = 0.0f; cB = NEG_INF; }
  float rS = cS, rB = cB;
#pragma unroll
  for (int j = 0; j < PER; ++j) {                   // inclusive per element
    rB = fmaxf(rB, rS + v[j].y);
    rS += v[j].x;
    base[lane * PER + j] = make_float2(rS, rB);
  }
}

// ---------------- kernel 3: recompute with carries, emit pulses ------------
__global__ void __launch_bounds__(BLOCK_T)
exgen_emit(const float* __restrict__ p, const float2* __restrict__ part,
           float* __restrict__ out) {
  __shared__ float ldsP[LDS_F];
  __shared__ float ldsPP[LDS_F];
  __shared__ float wTab[P_FRAME];
  const int chunk = blockIdx.x, row = blockIdx.y;
  const float* pRow = p + (size_t)row * N_FRAMES;
  stage_frames(pRow, chunk * FRAMES_PER_BLOCK, ldsP, ldsPP, wTab);

  const int lane = threadIdx.x & 31, wave = threadIdx.x >> 5;
  const float NEG_INF = -__builtin_inff();
  const int gw = chunk * WPB + wave;                // wave index within row
  float2 carry = (gw == 0) ? make_float2(0.0f, NEG_INF)
                           : part[(size_t)row * WAVES_PER_ROW + gw - 1];
  float runS    = carry.x;                          // s through previous sample
  float runBias = fmaxf(0.0f, carry.y);             // cummax(s*~mask) through prev
  float rPrev   = ceilf(runS - runBias);            // r[n-1] at tile entry
  float* outRow = out + (size_t)row * S_ROW + (size_t)chunk * BLOCK_TILE;

  const int base = wave * WAVE_TILE + lane;
  for (int s2 = 0; s2 < STEPS; ++s2) {
    int nLocal = base + s2 * 32;
    bool mask; float psamp, q;
    sample_q(ldsP, ldsPP, wTab, nLocal, mask, psamp, q);
    float sInc = wave_scan_add(q, lane);
    float sAbs = runS + sInc;
    float cand = mask ? 0.0f : sAbs;                // candidates all >= 0
    float bias = fmaxf(runBias, wave_scan_max(cand, lane));
    float phase = sAbs - bias;
    float r = ceilf(phase);
    float rUp = __shfl_up(r, 1, 32);
    if (lane == 0) rUp = rPrev;
    bool pulse = (r - rUp) >= 1.0f;
    float e = pulse ? sqrtf(psamp) : 0.0f;
    __builtin_nontemporal_store(e, outRow + nLocal); // stream, don't pollute L2
    runS    = bcast31(sAbs);
    runBias = bcast31(bias);
    rPrev   = bcast31(r);
  }
}

// ---------------- host-side launcher ---------------------------------------
extern "C" void kernel_launch(void* const* d_in, const int* in_sizes, int n_in,
                              void* d_out, int out_size, void* d_ws, size_t ws_size,
                              hipStream_t stream) {
  (void)in_sizes; (void)n_in; (void)out_size; (void)ws_size;
  const float* p = (const float*)d_in[0];
  float*  out  = (float*)d_out;
  float2* part = (float2*)d_ws;                     // 64*256*8 B = 128 KB scratch

  dim3 gridA(CHUNKS, B_ROWS);
  hipLaunchKernelGGL(exgen_partials, gridA, dim3(BLOCK_T), 0, stream, p, part);
  hipLaunchKernelGGL(exgen_scan, dim3(B_ROWS), dim3(32), 0, stream, part);
  hipLaunchKernelGGL(exgen_emit, gridA, dim3(BLOCK_T), 0, stream, p, part, out);
}